// QuantumConvolutionLayer_56289841381405
// MI455X (gfx1250) — compile-verified
//
#include <hip/hip_runtime.h>
#include <hip/hip_bf16.h>
#include <cstdint>

#define AS1 __attribute__((address_space(1)))
#define AS3 __attribute__((address_space(3)))
typedef int v4i __attribute__((ext_vector_type(4)));
typedef float v2f __attribute__((ext_vector_type(2)));
typedef float v8f __attribute__((ext_vector_type(8)));

// ---- gfx1250 async global->LDS copy (B128) ----
#if __has_builtin(__builtin_amdgcn_global_load_async_to_lds_b128)
#define ASYNC_COPY_B128(g, l, OFF)                                            \
  __builtin_amdgcn_global_load_async_to_lds_b128(                             \
      (AS1 v4i*)((g) + (OFF)), (AS3 v4i*)((l) + (OFF)), 0, 0)
#else
#define ASYNC_COPY_B128(g, l, OFF)                                            \
  asm volatile("global_load_async_to_lds_b128 %0, %1, off offset:" #OFF       \
               :: "v"((unsigned)(uintptr_t)((l) + (OFF))),                    \
                  "v"((g) + (OFF))                                            \
               : "memory")
#endif

#if __has_builtin(__builtin_amdgcn_s_wait_asynccnt)
#define WAIT_ASYNC(N) __builtin_amdgcn_s_wait_asynccnt(N)
#else
#define WAIT_ASYNC(N) asm volatile("s_wait_asynccnt " #N ::: "memory")
#endif

#if __has_builtin(__builtin_amdgcn_wmma_f32_16x16x4_f32)
#define HAVE_WMMA_F32_4 1
#else
#define HAVE_WMMA_F32_4 0
#endif

// ---------------------------------------------------------------------------
// Setup: compose the fixed 4x4 circuit matrix U from weights.
// Layer l: U_l = P_cnot * (RY(w[l,0]) (x) RY(w[l,1])); U = U_{L-1} ... U_0.
// ---------------------------------------------------------------------------
__global__ void quanv_setup_u(const float* __restrict__ w, float* __restrict__ U,
                              int n_layers) {
  if (threadIdx.x != 0 || blockIdx.x != 0) return;
  float M[16];
#pragma unroll
  for (int i = 0; i < 16; ++i) M[i] = (i % 5 == 0) ? 1.0f : 0.0f;  // identity

  for (int l = 0; l < n_layers; ++l) {
    float c0 = cosf(0.5f * w[2 * l + 0]);
    float s0 = sinf(0.5f * w[2 * l + 0]);
    float c1 = cosf(0.5f * w[2 * l + 1]);
    float s1 = sinf(0.5f * w[2 * l + 1]);
    float R0[4] = {c0, -s0, s0, c0};
    float R1[4] = {c1, -s1, s1, c1};
    float K[16];
    for (int a = 0; a < 2; ++a)
      for (int b = 0; b < 2; ++b)
        for (int ap = 0; ap < 2; ++ap)
          for (int bp = 0; bp < 2; ++bp)
            K[(2 * a + b) * 4 + (2 * ap + bp)] = R0[a * 2 + ap] * R1[b * 2 + bp];
    // CNOT(control=q0, target=q1): swap state components 2,3 -> swap rows 2,3
    for (int j = 0; j < 4; ++j) {
      float t = K[8 + j]; K[8 + j] = K[12 + j]; K[12 + j] = t;
    }
    float N[16];
    for (int i = 0; i < 4; ++i)
      for (int j = 0; j < 4; ++j) {
        float acc = 0.0f;
        for (int k = 0; k < 4; ++k) acc += K[i * 4 + k] * M[k * 4 + j];
        N[i * 4 + j] = acc;
      }
#pragma unroll
    for (int i = 0; i < 16; ++i) M[i] = N[i];
  }
#pragma unroll
  for (int i = 0; i < 16; ++i) U[i] = M[i];
}

// ---------------------------------------------------------------------------
// Main: one wave32 per output row (128 px). Wave-private double-buffered
// async global->LDS pipeline over even input rows. Per 16-pixel batch, the
// 4x4 circuit matvec runs on the matrix pipe via V_WMMA_F32_16X16X4_F32:
//   A (16x4)  = U replicated 4x along M  (per-lane: 2 floats, loaded once)
//   B (4x16)  = e-vectors; lanes 0-15 supply c0*(c1,s1) (K=0,1),
//               lanes 16-31 supply s0*(c1,s1) (K=2,3) -- layout-native, no
//               cross-lane shuffles needed.
//   D gives y0..y3 of pixel (lane&15) in d[0..3] on every lane; lanes 0-15
//   store the <Z0> plane, lanes 16-31 the <Z1> plane.
// ---------------------------------------------------------------------------
__global__ __launch_bounds__(256) void quanv_main(
    const float* __restrict__ x, const float* __restrict__ Uc,
    float* __restrict__ out, int nRows, int waveStride) {
  __shared__ char lds[8 * 2048];  // 8 waves * (2 buffers * 1KB)
  const int lane = threadIdx.x & 31;
  const int wib = threadIdx.x >> 5;
  const int n = lane & 15;    // pixel-in-batch / output column (mod 16)
  const int half = lane >> 4; // 0: cos(t0)/Z0 half, 1: sin(t0)/Z1 half
  char* myLds = lds + wib * 2048;
  const float HPI = 1.57079632679489662f;  // pi/2

#if HAVE_WMMA_F32_4
  v2f A;  // A-matrix fragment: row (lane&3) of U, K-pair (half*2, half*2+1)
  A[0] = Uc[(lane & 3) * 4 + half * 2 + 0];
  A[1] = Uc[(lane & 3) * 4 + half * 2 + 1];
#else
  float u[16];
#pragma unroll
  for (int i = 0; i < 16; ++i) u[i] = Uc[i];
#endif

  int g = blockIdx.x * 8 + wib;  // global row id: g = plane*128 + h

  if (g < nRows) {  // prologue: prefetch first row into buffer 0
    int plane = g >> 7, h = g & 127;
    const char* gb = (const char*)(x + ((size_t)plane * 256 + 2 * h) * 256) +
                     lane * 16;
    char* lb = myLds + lane * 16;
    ASYNC_COPY_B128(gb, lb, 0);
    ASYNC_COPY_B128(gb, lb, 512);
  }

  int stage = 0;
  for (; g < nRows; g += waveStride, stage ^= 1) {
    int gn = g + waveStride;
    if (gn < nRows) {  // prefetch next row into the other buffer
      int plane = gn >> 7, h = gn & 127;
      const char* gb = (const char*)(x + ((size_t)plane * 256 + 2 * h) * 256) +
                       lane * 16;
      char* lb = myLds + (stage ^ 1) * 1024 + lane * 16;
      ASYNC_COPY_B128(gb, lb, 0);
      ASYNC_COPY_B128(gb, lb, 512);
      WAIT_ASYNC(2);  // async loads complete in order -> current row resident
    } else {
      WAIT_ASYNC(0);
    }

    const float* lrow = (const float*)(myLds + stage * 1024);
    int plane = g >> 7, h = g & 127;
    float* z0p = out + ((size_t)(2 * plane) * 128 + h) * 128;  // channel 2c
    float* zp = z0p + (size_t)half * (128 * 128);  // this half's plane

#pragma unroll
    for (int t = 0; t < 8; ++t) {
      // pixel p = 16t + n: (x0, x1) pair; both half-waves read same address
      float2 px = *(const float2*)(lrow + (t * 16 + n) * 2);
      float s1, c1;
      __sincosf(HPI * px.y, &s1, &c1);
      float y0, y1, y2, y3;
#if HAVE_WMMA_F32_4
      // a = cos(t0) on lanes 0-15, cos(t0 - pi/2) = sin(t0) on lanes 16-31
      float a = __cosf(HPI * px.x - (float)half * HPI);
      v2f B;
      B[0] = a * c1;
      B[1] = a * s1;
      v8f c = {};
      v8f d = __builtin_amdgcn_wmma_f32_16x16x4_f32(
          false, A, false, B, (short)0, c, false, false);
      y0 = d[0]; y1 = d[1]; y2 = d[2]; y3 = d[3];
#else
      float s0, c0;
      __sincosf(HPI * px.x, &s0, &c0);
      float e0 = c0 * c1, e1 = c0 * s1, e2 = s0 * c1, e3 = s0 * s1;
      y0 = fmaf(u[0],  e0, fmaf(u[1],  e1, fmaf(u[2],  e2, u[3]  * e3)));
      y1 = fmaf(u[4],  e0, fmaf(u[5],  e1, fmaf(u[6],  e2, u[7]  * e3)));
      y2 = fmaf(u[8],  e0, fmaf(u[9],  e1, fmaf(u[10], e2, u[11] * e3)));
      y3 = fmaf(u[12], e0, fmaf(u[13], e1, fmaf(u[14], e2, u[15] * e3)));
#endif
      float p0 = y0 * y0, p1 = y1 * y1, p2 = y2 * y2, p3 = y3 * y3;
      float z = half ? (p0 + p2) - (p1 + p3)   // <Z1>
                     : (p0 + p1) - (p2 + p3);  // <Z0>
      zp[t * 16 + n] = z;
    }
  }
}

extern "C" void kernel_launch(void* const* d_in, const int* in_sizes, int n_in,
                              void* d_out, int out_size, void* d_ws,
                              size_t ws_size, hipStream_t stream) {
  const float* x = (const float*)d_in[0];
  const float* w = (const float*)d_in[1];
  float* out = (float*)d_out;
  float* U = (float*)d_ws;  // 16 floats of scratch

  const int C = 3, H = 256, W = 256;
  int B = in_sizes[0] / (C * H * W);
  int n_layers = in_sizes[1] / 2;
  int nRows = B * C * (H / 2);  // one wave per output row

  quanv_setup_u<<<1, 1, 0, stream>>>(w, U, n_layers);

  int blocks = 1024;  // 8192 waves; 24576 rows -> 3 pipelined steps per wave
  if (blocks * 8 > nRows) blocks = (nRows + 7) / 8;
  quanv_main<<<blocks, 256, 0, stream>>>(x, U, out, nRows, blocks * 8);
}